// ASIC_51694226374850
// MI455X (gfx1250) — compile-verified
//
#include <hip/hip_runtime.h>
#include <cstdint>

#define D1 128
#define D2 128
#define XC 64
#define NB 32
#define NM 16
#define ROWS 2          // output rows per block
#define BPG 4           // batches per block
#define THREADS 256     // 8 wave32

typedef unsigned int u32x4 __attribute__((ext_vector_type(4)));
typedef int          i32x4 __attribute__((ext_vector_type(4)));
typedef int          i32x8 __attribute__((ext_vector_type(8)));

// Low 32 bits of a generic pointer into LDS == LDS byte offset
// (shared aperture occupies only the high 32 bits on gfx1250).
__device__ __forceinline__ uint32_t lds_off_of(const void* p) {
    return (uint32_t)(uintptr_t)p;
}

// CDNA5 async global->LDS DMA (ASYNCcnt-tracked), per-lane addresses.
__device__ __forceinline__ void async_f32_to_lds(uint32_t lds_byte_off, const float* gsrc) {
    asm volatile("global_load_async_to_lds_b32 %0, %1, off"
                 :: "v"(lds_byte_off), "v"((uint64_t)(uintptr_t)gsrc)
                 : "memory");
}

__device__ __forceinline__ void wait_asynccnt0() {
    asm volatile("s_wait_asynccnt 0x0" ::: "memory");
}

// CDNA5 Tensor Data Mover: one-descriptor 2D tile DMA global->LDS.
// tile_d1 rows of tile_d0 4-byte elements, row stride d0_stride elements.
__device__ __forceinline__ void tdm_load_2d_f32_to_lds(uint32_t lds_byte, const void* gaddr,
                                                       uint32_t tile_d0, uint32_t tile_d1,
                                                       uint64_t d0_stride) {
    const uint64_t ga = (uint64_t)(uintptr_t)gaddr;
    u32x4 g0;
    g0.x = 1u;                                               // count=1 (valid), user mode
    g0.y = lds_byte;                                         // lds_addr [63:32]
    g0.z = (uint32_t)ga;                                     // global_addr [95:64]
    g0.w = (uint32_t)((ga >> 32) & 0x01FFFFFFu) | (2u << 30);// global_addr[56:32] | type=2
    i32x8 g1;
    g1[0] = (int)(2u << 16);                                 // data_size=2 (4B); wg_mask=0
    g1[1] = (int)((tile_d0 & 0xFFFFu) << 16);                // tensor_dim0[15:0] (=tile_d0)
    g1[2] = (int)(((tile_d0 >> 16) & 0xFFFFu)                // tensor_dim0[31:16]
                  | ((tile_d1 & 0xFFFFu) << 16));            // tensor_dim1[15:0]
    g1[3] = (int)(((tile_d1 >> 16) & 0xFFFFu)                // tensor_dim1[31:16]
                  | ((tile_d0 & 0xFFFFu) << 16));            // tile_dim0
    g1[4] = (int)(tile_d1 & 0xFFFFu);                        // tile_dim1; tile_dim2=0
    g1[5] = (int)(uint32_t)d0_stride;                        // tensor_dim0_stride[31:0]
    g1[6] = (int)(uint32_t)((d0_stride >> 32) & 0xFFFFu);    // stride[47:32]; dim1_stride=0
    g1[7] = 0;
    const i32x4 z4 = {0, 0, 0, 0};
#if __clang_major__ >= 23
    const i32x8 z8 = {0, 0, 0, 0, 0, 0, 0, 0};
    __builtin_amdgcn_tensor_load_to_lds(g0, g1, z4, z4, z8, 0);
#else
    __builtin_amdgcn_tensor_load_to_lds(g0, g1, z4, z4, 0);
#endif
}

__global__ void __launch_bounds__(THREADS)
lca_layer(const float* __restrict__ xin,   // layer-0 input (first==1)
          const float* __restrict__ sin,   // state input  (first==0)
          const float* __restrict__ tg,    // toggle_gates (L,16,128,128) raw
          float* __restrict__ sout,        // state output (last==0)
          float* __restrict__ fout,        // final output (last==1), (B,128,64)
          int layer, int first, int last)
{
    __shared__ float tw_s[NM][ROWS][D2];        // 16 KB: sigmoided gate tile
    __shared__ float rows_s[2][ROWS + 1][D2];   // 3 KB: double-buffered state rows

    const int t  = threadIdx.x;
    const int r0 = blockIdx.x * ROWS;           // first output row of this block
    const int b0 = blockIdx.y * BPG;            // first batch of this block

    float* tw_flat = &tw_s[0][0][0];
    const uint32_t tw_base   = lds_off_of(tw_flat);
    const uint32_t rows_base = lds_off_of(&rows_s[0][0][0]);
    const uint32_t buf_bytes = (ROWS + 1) * D2 * 4u;

    // ---- gate tile (16 x (2*128) f32, row stride 128*128) via one TDM descriptor ----
    if (t < 32) {   // wave 0 only; TENSORcnt-tracked
        const float* tile = tg + (size_t)layer * NM * D1 * D2 + (size_t)r0 * D2;
        tdm_load_2d_f32_to_lds(tw_base, tile, ROWS * D2, NM, (uint64_t)D1 * D2);
    }

    // ---- stage state rows for batch b into LDS buffer sel ----
    auto stage_rows = [&](int b, int sel) {
        if (first) {
            // embed: even grid columns hold x, odd columns are zero
            for (int idx = t; idx < (ROWS + 1) * D2; idx += THREADS) {
                const int r = idx >> 7, cc = idx & (D2 - 1);
                const int row = (r0 + r) & (D1 - 1);
                float v = 0.0f;
                if (!(cc & 1)) v = xin[((size_t)b * D1 + row) * XC + (cc >> 1)];
                rows_s[sel][r][cc] = v;
            }
        } else {
            for (int idx = t; idx < (ROWS + 1) * D2; idx += THREADS) {
                const int r = idx >> 7, cc = idx & (D2 - 1);
                const int row = (r0 + r) & (D1 - 1);
                async_f32_to_lds(rows_base + (uint32_t)sel * buf_bytes + (uint32_t)idx * 4u,
                                 sin + ((size_t)b * D1 + row) * D2 + cc);
            }
        }
    };

    // prologue: overlap first batch's row DMA with the gate-tile TDM transfer
    stage_rows(b0, 0);
    wait_asynccnt0();
    if (t < 32) __builtin_amdgcn_s_wait_tensorcnt(0);
    __syncthreads();

    // sigmoid in place (batch-invariant: computed once, reused by all batches)
    #pragma unroll
    for (int i = 0; i < (NM * ROWS * D2) / THREADS; ++i) {
        const int idx = i * THREADS + t;
        const float v = tw_flat[idx];
        tw_flat[idx] = 1.0f / (1.0f + __expf(-v));
    }
    __syncthreads();

    const int rr = t >> 7;                      // 0..1: which output row
    const int c  = t & (D2 - 1);                // column
    const int p1 = r0 + rr;                     // output row index

    for (int bi = 0; bi < BPG; ++bi) {
        const int b   = b0 + bi;
        const int cur = bi & 1;

        // software pipeline: kick off next batch's DMA before computing this one
        if (bi + 1 < BPG) stage_rows(b + 1, cur ^ 1);

        // ---- 2x2 taps (c0..c3), then 15-FMA multilinear lerp over tw ----
        const int cp = (c + 1) & (D2 - 1);
        const float s00 = rows_s[cur][rr][c];        // (i1=0,i2=0) -> bit 0 (MSB of m)
        const float s01 = rows_s[cur][rr][cp];       // (0,1) -> bit 1
        const float s10 = rows_s[cur][rr + 1][c];    // (1,0) -> bit 2
        const float s11 = rows_s[cur][rr + 1][cp];   // (1,1) -> bit 3 (LSB of m)

        float g[8];
        #pragma unroll
        for (int i = 0; i < 8; ++i) {                // reduce over bit3 with s11
            const float a  = tw_s[2 * i][rr][c];
            const float bb = tw_s[2 * i + 1][rr][c];
            g[i] = fmaf(s11, bb - a, a);
        }
        float h[4];
        #pragma unroll
        for (int i = 0; i < 4; ++i)                  // reduce over bit2 with s10
            h[i] = fmaf(s10, g[2 * i + 1] - g[2 * i], g[2 * i]);
        const float k0 = fmaf(s01, h[1] - h[0], h[0]);   // bit1 with s01
        const float k1 = fmaf(s01, h[3] - h[2], h[2]);
        float v = fmaf(s00, k1 - k0, k0);                // bit0 with s00
        v = fminf(fmaxf(v, 0.0f), 1.0f);

        if (last) {
            if (!(c & 1))
                fout[((size_t)b * D1 + p1) * XC + (c >> 1)] = v;
        } else {
            sout[((size_t)b * D1 + p1) * D2 + c] = v;
        }

        // make next buffer visible to all waves; also fences reads of `cur`
        // before it is overwritten in the next iteration
        if (bi + 1 < BPG) {
            wait_asynccnt0();
            __syncthreads();
        }
    }
}

extern "C" void kernel_launch(void* const* d_in, const int* in_sizes, int n_in,
                              void* d_out, int out_size, void* d_ws, size_t ws_size,
                              hipStream_t stream) {
    const float* x  = (const float*)d_in[0];   // (32,128,64)
    const float* tg = (const float*)d_in[1];   // (4,16,128,128)
    float* out  = (float*)d_out;               // (32,128,64)
    float* buf0 = (float*)d_ws;                // (32,128,128) state ping
    float* buf1 = buf0 + (size_t)NB * D1 * D2; // (32,128,128) state pong

    dim3 grid(D1 / ROWS, NB / BPG);            // 64 x 8 blocks
    dim3 block(THREADS);

    lca_layer<<<grid, block, 0, stream>>>(x, nullptr, tg, buf0, nullptr, 0, 1, 0);
    lca_layer<<<grid, block, 0, stream>>>(nullptr, buf0, tg, buf1, nullptr, 1, 0, 0);
    lca_layer<<<grid, block, 0, stream>>>(nullptr, buf1, tg, buf0, nullptr, 2, 0, 0);
    lca_layer<<<grid, block, 0, stream>>>(nullptr, buf0, tg, nullptr, out, 3, 0, 1);
}